// WeightedAttentionPooling_comp_53274774340081
// MI455X (gfx1250) — compile-verified
//
#include <hip/hip_runtime.h>
#include <hip/hip_bf16.h>

// ---------------------------------------------------------------------------
// WeightedAttentionPooling on MI455X (gfx1250, wave32)
//
// out[s] = ( sum_{i in s} alpha_i * x_i / sum_{i in s} alpha_i ) @ Wm + bm
// with alpha_i = pos_i * exp(x_i . Wg + bg)   (softmax weights are linear in
// the final sum, so the N x D x D GEMM collapses to an S x D x D GEMM).
//
// Phase 1: stream x once (512 MB => ~22us @ 23.3 TB/s, the roofline floor),
//          accumulate per-segment Y = sum alpha*x and asum = sum alpha using
//          run-length accumulation over the *sorted* batch_index, flushing
//          with global_atomic_add_f32 only at segment boundaries.
// Phase 2: f32 WMMA (v_wmma_f32_16x16x4_f32) GEMM: (Y/asum)[S,128] @ Wm[128,128]
//          + bm, A tiles staged in LDS, 8 col-tiles per block (1 wave each).
// ---------------------------------------------------------------------------

typedef __attribute__((ext_vector_type(2))) float v2f;
typedef __attribute__((ext_vector_type(8))) float v8f;

#define DIM 128
#define NODES_PER_WAVE 64

// ---------------------------------------------------------------------------
// Phase 1: fused gate + alpha + per-segment weighted accumulation.
// One wave handles NODES_PER_WAVE consecutive nodes; each lane owns 4 of the
// 128 features (float4). Dot product reduced with a 5-step xor butterfly.
// ---------------------------------------------------------------------------
__global__ void __launch_bounds__(256)
node_pass(const float* __restrict__ x, const float* __restrict__ pos,
          const int* __restrict__ batch, const float* __restrict__ Wg,
          const float* __restrict__ bg, float* __restrict__ Y,
          float* __restrict__ asum, int N) {
  const int gwave = (int)((blockIdx.x * (unsigned)blockDim.x + threadIdx.x) >> 5);
  const int lane  = threadIdx.x & 31;

  int n0 = gwave * NODES_PER_WAVE;
  if (n0 >= N) return;
  int n1 = n0 + NODES_PER_WAVE;
  if (n1 > N) n1 = N;

  const float4 wg  = reinterpret_cast<const float4*>(Wg)[lane]; // Wg[4*lane..4*lane+3]
  const float  bg0 = bg[0];

  float acc0 = 0.f, acc1 = 0.f, acc2 = 0.f, acc3 = 0.f, accA = 0.f;
  int cur = batch[n0];

  for (int n = n0; n < n1; ++n) {
    const int seg = batch[n];
    if (seg != cur) {
      float* yrow = Y + (size_t)cur * DIM + 4 * lane;
      unsafeAtomicAdd(yrow + 0, acc0);
      unsafeAtomicAdd(yrow + 1, acc1);
      unsafeAtomicAdd(yrow + 2, acc2);
      unsafeAtomicAdd(yrow + 3, acc3);
      if (lane == 0) unsafeAtomicAdd(&asum[cur], accA);
      acc0 = acc1 = acc2 = acc3 = accA = 0.f;
      cur = seg;
    }
    const float4 xv = reinterpret_cast<const float4*>(x + (size_t)n * DIM)[lane];
    float d = xv.x * wg.x + xv.y * wg.y + xv.z * wg.z + xv.w * wg.w;
    d += __shfl_xor(d, 16, 32);
    d += __shfl_xor(d, 8, 32);
    d += __shfl_xor(d, 4, 32);
    d += __shfl_xor(d, 2, 32);
    d += __shfl_xor(d, 1, 32);
    const float alpha = pos[n] * __expf(d + bg0);
    accA += alpha;
    acc0 += alpha * xv.x;
    acc1 += alpha * xv.y;
    acc2 += alpha * xv.z;
    acc3 += alpha * xv.w;
  }
  {
    float* yrow = Y + (size_t)cur * DIM + 4 * lane;
    unsafeAtomicAdd(yrow + 0, acc0);
    unsafeAtomicAdd(yrow + 1, acc1);
    unsafeAtomicAdd(yrow + 2, acc2);
    unsafeAtomicAdd(yrow + 3, acc3);
    if (lane == 0) unsafeAtomicAdd(&asum[cur], accA);
  }
}

// ---------------------------------------------------------------------------
// Phase 2: out[16 rows x 128 cols per block] = (Y/asum) @ Wm + bm via
// v_wmma_f32_16x16x4_f32. 256 threads = 8 waves; wave w owns col tile w (16
// cols), all waves share the normalized 16x128 A tile in LDS.
//
// ISA VGPR layouts (CDNA5 ISA):
//   A (16x4 f32):  v0 = K=0 (lanes 0-15) / K=2 (lanes 16-31); v1 = K=1 / K=3,
//                  M = lane % 16.
//   B (4x16 f32):  transposed analog: v0 = row K=0 / K=2 striped over lanes
//                  (N = lane % 16), v1 = row K=1 / K=3.
//   C/D (16x16 f32): vr = M=r (lanes 0-15) / M=r+8 (lanes 16-31), N = lane%16.
// ---------------------------------------------------------------------------
__global__ void __launch_bounds__(256)
seg_gemm(const float* __restrict__ Y, const float* __restrict__ asum,
         const float* __restrict__ Wm, const float* __restrict__ bm,
         float* __restrict__ out, int S) {
  __shared__ float As[16][DIM + 2];   // +2 pad (even) to dodge bank conflicts

  const int seg0 = blockIdx.x * 16;
  const int tid  = threadIdx.x;

  // Cooperative load + normalize of the A tile (16 segments x 128 features).
  for (int i = tid; i < 16 * DIM; i += 256) {
    const int r = i >> 7, c = i & (DIM - 1);
    const int s = seg0 + r;
    float v = 0.f;
    if (s < S) {
      const float a = asum[s];
      if (a != 0.f) v = Y[(size_t)s * DIM + c] / a;
    }
    As[r][c] = v;
  }
  __syncthreads();

  const int wave = tid >> 5;
  const int lane = tid & 31;
  const int half = lane >> 4;       // 0: lanes 0-15, 1: lanes 16-31
  const int lr   = lane & 15;
  const int col0 = wave * 16;

  v8f c = {};
  #pragma unroll
  for (int k0 = 0; k0 < DIM; k0 += 4) {
    const int ka = k0 + 2 * half;
    v2f a, b;
    a[0] = As[lr][ka];
    a[1] = As[lr][ka + 1];
    b[0] = Wm[(size_t)ka * DIM + col0 + lr];
    b[1] = Wm[(size_t)(ka + 1) * DIM + col0 + lr];
    // (neg_a, A, neg_b, B, c_mod, C, reuse_a, reuse_b)
    c = __builtin_amdgcn_wmma_f32_16x16x4_f32(false, a, false, b,
                                              (short)0, c, false, false);
  }

  const float bias = bm[col0 + lr];
  #pragma unroll
  for (int r = 0; r < 8; ++r) {
    const int s = seg0 + r + 8 * half;
    if (s < S) out[(size_t)s * DIM + col0 + lr] = c[r] + bias;
  }
}

// ---------------------------------------------------------------------------
// Inputs (setup_inputs order): 0:x[N,128] f32, 1:edge_index (unused),
// 2:pos[N] f32, 3:batch_index[N] i32, 4:Wg[128] f32, 5:bg[1] f32,
// 6:Wm[128,128] f32, 7:bm[128] f32.  Output: out[S,128] f32.
// ---------------------------------------------------------------------------
extern "C" void kernel_launch(void* const* d_in, const int* in_sizes, int n_in,
                              void* d_out, int out_size, void* d_ws, size_t ws_size,
                              hipStream_t stream) {
  const float* x     = (const float*)d_in[0];
  const float* pos   = (const float*)d_in[2];
  const int*   batch = (const int*)d_in[3];
  const float* Wg    = (const float*)d_in[4];
  const float* bg    = (const float*)d_in[5];
  const float* Wm    = (const float*)d_in[6];
  const float* bm    = (const float*)d_in[7];
  float*       out   = (float*)d_out;

  const int N = in_sizes[2];          // pos has N elements
  const int S = out_size / DIM;

  float* Y    = (float*)d_ws;                 // S*DIM floats
  float* asum = Y + (size_t)S * DIM;          // S floats

  hipMemsetAsync(d_ws, 0, ((size_t)S * DIM + (size_t)S) * sizeof(float), stream);

  const int waves   = (N + NODES_PER_WAVE - 1) / NODES_PER_WAVE;
  const int blocks1 = (waves + 7) / 8;        // 8 waves (256 threads) per block
  node_pass<<<blocks1, 256, 0, stream>>>(x, pos, batch, Wg, bg, Y, asum, N);

  const int blocks2 = (S + 15) / 16;          // 16 segment rows per block
  seg_gemm<<<blocks2, 256, 0, stream>>>(Y, asum, Wm, bm, out, S);
}